// SAGEConv_40123584479253
// MI455X (gfx1250) — compile-verified
//
#include <hip/hip_runtime.h>

typedef __attribute__((ext_vector_type(2))) float v2f;
typedef __attribute__((ext_vector_type(8))) float v8f;
typedef __attribute__((ext_vector_type(4))) int   v4i;

#define N_NODES 100000
#define N_EDGES 640000
#define D 128
#define ROWS 32          // node rows per block (two 16-row WMMA tiles)
#define LDS_PITCH 132    // 128 + 4 pad: conflict-free column reads; rows stay 16B-aligned

#if __has_builtin(__builtin_amdgcn_global_load_async_to_lds_b128)
#define HAVE_ASYNC_LDS 1
#else
#define HAVE_ASYNC_LDS 0
#endif

typedef __attribute__((address_space(1))) v4i gv4i;   // global int4
typedef __attribute__((address_space(3))) v4i lv4i;   // LDS int4

__device__ __forceinline__ void wait_asynccnt0() {
#if __has_builtin(__builtin_amdgcn_s_wait_asynccnt)
    __builtin_amdgcn_s_wait_asynccnt(0);
#else
    asm volatile("s_wait_asynccnt 0x0" ::: "memory");
#endif
}

// ---------------------------------------------------------------------------
// Kernel 1: zero the scatter accumulators in workspace (grid-stride).
// ---------------------------------------------------------------------------
__global__ void sage_zero(float* __restrict__ ws, int n) {
    int i = blockIdx.x * blockDim.x + threadIdx.x;
    int stride = gridDim.x * blockDim.x;
    for (; i < n; i += stride) ws[i] = 0.0f;
}

// ---------------------------------------------------------------------------
// Kernel 2: pack W (row-major 128x128) into K-pair-major float2 layout:
//   Wp[p = k/2][col] = { W[2p][col], W[2p+1][col] }
// so a WMMA B-fragment is ONE coalesced global_load_b64 per lane.
// ---------------------------------------------------------------------------
__global__ void sage_pack_w(const float* __restrict__ Wself,
                            const float* __restrict__ Wneigh,
                            float2* __restrict__ WpSelf,
                            float2* __restrict__ WpNeigh) {
    int t = blockIdx.x * blockDim.x + threadIdx.x;   // 0 .. 2*64*128-1
    int m = t >> 13;                                 // matrix select
    int p = t & 8191;                                // pair index
    int kp  = p >> 7;                                // k/2
    int col = p & 127;
    const float* W  = m ? Wneigh : Wself;
    float2*      Wp = m ? WpNeigh : WpSelf;
    float2 v;
    v.x = W[(2 * kp) * D + col];
    v.y = W[(2 * kp + 1) * D + col];
    Wp[p] = v;
}

// ---------------------------------------------------------------------------
// Kernel 3: edge scatter. One wave32 per edge; each lane handles 4 contiguous
// floats (float4 gather -> 4 global_atomic_add_f32). Lane 0 bumps the degree.
// Accumulator (51 MB) is L2-resident (192 MB L2), so the RMW stays on-chip.
// ---------------------------------------------------------------------------
__global__ __launch_bounds__(256) void sage_scatter(
        const float* __restrict__ x,
        const int*   __restrict__ src,
        const int*   __restrict__ dst,
        float* __restrict__ nsum,
        float* __restrict__ ncnt) {
    unsigned e    = (blockIdx.x * blockDim.x + threadIdx.x) >> 5;  // edge id
    unsigned lane = threadIdx.x & 31;
    if (e >= N_EDGES) return;
    int s = src[e];
    int d = dst[e];
    const float4 v = *(const float4*)(x + (size_t)s * D + lane * 4);
    float* o = nsum + (size_t)d * D + lane * 4;
    atomicAdd(o + 0, v.x);
    atomicAdd(o + 1, v.y);
    atomicAdd(o + 2, v.z);
    atomicAdd(o + 3, v.w);
    if (lane == 0) atomicAdd(ncnt + d, 1.0f);
}

// ---------------------------------------------------------------------------
// Kernel 4: fused mean + dual fp32 WMMA GEMM + bias.
//   32 node rows x 128 cols per block; 8 waves, wave w -> cols [16w,16w+16),
//   each wave owns TWO 16x16 C tiles (rows 0-15 / 16-31) sharing B fragments.
//   x-tile and RAW nsum-tile staged via async global->LDS (ASYNCcnt path);
//   the 1/degree scaling is applied to the neighbor A-fragments in-register
//   (diag(inv)*S @ W == (inv_r * S_frag) @ W), so the LDS copy is a pure DMA.
// Operand layout (ISA 7.12.2, 32-bit, wave32):
//   A: lane<16 -> row M=lane, {K=k0,k0+1}; lane>=16 -> {K=k0+2,k0+3}
//   B: lane<16 -> col N=lane, {K=k0,k0+1}; lane>=16 -> {K=k0+2,k0+3}
//   C/D: VGPR i -> M = i + 8*(lane>>4), N = (lane&15) + 16*wave
// ---------------------------------------------------------------------------
__global__ __launch_bounds__(256) void sage_gemm(
        const float*  __restrict__ x,
        const float2* __restrict__ WpSelf,  const float* __restrict__ bself,
        const float2* __restrict__ WpNeigh, const float* __restrict__ bneigh,
        const float*  __restrict__ nsum,    const float* __restrict__ ncnt,
        float* __restrict__ out) {
    __shared__ float ldsX[ROWS * LDS_PITCH];   // x tile, raw
    __shared__ float ldsS[ROWS * LDS_PITCH];   // neighbor-sum tile, raw

    const int tid   = threadIdx.x;
    const int node0 = blockIdx.x * ROWS;

    // Stage both 32x128 fp32 tiles into LDS, 16B granularity.
    for (int idx = tid; idx < ROWS * 32; idx += 256) {
        int row = idx >> 5;
        int c4  = (idx & 31) * 4;
        size_t g = (size_t)(node0 + row) * D + c4;
        int    l = row * LDS_PITCH + c4;
#if HAVE_ASYNC_LDS
        __builtin_amdgcn_global_load_async_to_lds_b128(
            (gv4i*)(x + g),    (lv4i*)&ldsX[l], 0, 0);
        __builtin_amdgcn_global_load_async_to_lds_b128(
            (gv4i*)(nsum + g), (lv4i*)&ldsS[l], 0, 0);
#else
        *(float4*)&ldsX[l] = *(const float4*)(x + g);
        *(float4*)&ldsS[l] = *(const float4*)(nsum + g);
#endif
    }
#if HAVE_ASYNC_LDS
    wait_asynccnt0();
#endif
    __syncthreads();

    const int wave = tid >> 5;
    const int lane = tid & 31;
    const int half = lane >> 4;
    const int r    = lane & 15;
    const int col  = wave * 16 + r;       // output column owned by this lane

    // Per-lane mean reciprocals for the two rows this lane feeds (A rows r, r+16).
    float invA = 1.0f / fmaxf(ncnt[node0 + r], 1.0f);
    float invB = 1.0f / fmaxf(ncnt[node0 + 16 + r], 1.0f);

    float bias = bself[col] + bneigh[col];
    v8f acc0 = {bias, bias, bias, bias, bias, bias, bias, bias};  // rows 0-15
    v8f acc1 = acc0;                                              // rows 16-31

    // out += x @ W_self   (B fragment: single b64 from packed Wp)
    #pragma unroll 4
    for (int k0 = 0; k0 < D; k0 += 4) {
        int k = k0 + 2 * half;
        v2f b  = *(const v2f*)(&WpSelf[(k >> 1) * D + col]);
        v2f a0 = *(const v2f*)(&ldsX[r * LDS_PITCH + k]);
        v2f a1 = *(const v2f*)(&ldsX[(r + 16) * LDS_PITCH + k]);
        acc0 = __builtin_amdgcn_wmma_f32_16x16x4_f32(
                   false, a0, false, b, (short)0, acc0, false, false);
        acc1 = __builtin_amdgcn_wmma_f32_16x16x4_f32(
                   false, a1, false, b, (short)0, acc1, false, false);
    }
    // out += (diag(inv) * S) @ W_neigh, scaling folded into A fragments
    #pragma unroll 4
    for (int k0 = 0; k0 < D; k0 += 4) {
        int k = k0 + 2 * half;
        v2f b  = *(const v2f*)(&WpNeigh[(k >> 1) * D + col]);
        v2f a0 = *(const v2f*)(&ldsS[r * LDS_PITCH + k]) * invA;
        v2f a1 = *(const v2f*)(&ldsS[(r + 16) * LDS_PITCH + k]) * invB;
        acc0 = __builtin_amdgcn_wmma_f32_16x16x4_f32(
                   false, a0, false, b, (short)0, acc0, false, false);
        acc1 = __builtin_amdgcn_wmma_f32_16x16x4_f32(
                   false, a1, false, b, (short)0, acc1, false, false);
    }

    // Write back: VGPR i of tile0 -> row i + 8*half; tile1 -> +16.
    #pragma unroll
    for (int i = 0; i < 8; i++) {
        int row = i + 8 * half;
        out[(size_t)(node0 + row) * D + col]      = acc0[i];
        out[(size_t)(node0 + 16 + row) * D + col] = acc1[i];
    }
}

// ---------------------------------------------------------------------------
extern "C" void kernel_launch(void* const* d_in, const int* in_sizes, int n_in,
                              void* d_out, int out_size, void* d_ws, size_t ws_size,
                              hipStream_t stream) {
    const float* x      = (const float*)d_in[0];
    const int*   edges  = (const int*)d_in[1];         // [2, N_EDGES] flat
    const float* Wself  = (const float*)d_in[2];
    const float* bself  = (const float*)d_in[3];
    const float* Wneigh = (const float*)d_in[4];
    const float* bneigh = (const float*)d_in[5];
    float*       out    = (float*)d_out;

    const int* src = edges;
    const int* dst = edges + N_EDGES;

    // Workspace layout (fp32): nsum[N*D] | ncnt[N] | WpSelf[8192 f2] | WpNeigh[8192 f2]
    float*  nsum    = (float*)d_ws;
    float*  ncnt    = nsum + (size_t)N_NODES * D;
    float2* WpSelf  = (float2*)(ncnt + N_NODES);       // 8-byte aligned
    float2* WpNeigh = WpSelf + 64 * D;

    // 1) zero accumulators (re-run every call: graph replay safe)
    int zero_n = N_NODES * D + N_NODES;
    sage_zero<<<2048, 256, 0, stream>>>(nsum, zero_n);

    // 2) pack both weight matrices into K-pair-major float2 (16384 threads)
    sage_pack_w<<<64, 256, 0, stream>>>(Wself, Wneigh, WpSelf, WpNeigh);

    // 3) scatter-add: 8 edges per 256-thread block
    sage_scatter<<<N_EDGES / 8, 256, 0, stream>>>(x, src, dst, nsum, ncnt);

    // 4) fused mean + dual GEMM + bias: 32 nodes per block (100000/32 = 3125)
    sage_gemm<<<N_NODES / ROWS, 256, 0, stream>>>(x, WpSelf, bself, WpNeigh, bneigh,
                                                  nsum, ncnt, out);
}